// SimilarityMeasureModel_57389353009457
// MI455X (gfx1250) — compile-verified
//
#include <hip/hip_runtime.h>
#include <stdint.h>

// HorizontalComparisonModel: per (b,p,n) pair reduce over C=512:
//   l1 = sum|x-y|, l2 = sum (x-y)^2, output [l1, l2, l1] per pair.
// Bandwidth-bound: ~403 MB streamed @ 23.3 TB/s => ~17 us floor; 0.5 FLOP/B,
// so WMMA cannot help. CDNA5 paths used: async global->LDS DMA FIFO
// (global_load_async_to_lds_b128 / ASYNCcnt) + split wait counters.

namespace {
constexpr int kB  = 32;
constexpr int kP  = 3;
constexpr int kC  = 512;
constexpr int kN  = 1024;
constexpr int kBP = kB * kP;            // 96 slices
constexpr int kLanes = 32;              // wave32
constexpr int kCW    = 8;               // 8 waves per WG, wave w owns c == w (mod 8)
constexpr int kSteps = kC / kCW;        // 64 c-steps per wave
constexpr int kDepth = 8;               // async FIFO depth (slots per wave)
constexpr int kTileB = kLanes * 16;     // 512 B: one contiguous row-tile (32 lanes x float4)
constexpr int kSlotB = 2 * kTileB;      // 1 KB per slot: x-tile + y-tile
constexpr int kWaveLds = kDepth * kSlotB;   // 8 KB per wave
constexpr int kWgLds   = kCW * kWaveLds;    // 64 KB per workgroup
constexpr int kOutPerB = kP * kN * 3;       // 9216 floats per batch row
}

__global__ __launch_bounds__(kLanes * kCW)
void horiz_cmp_kernel(const float* __restrict__ x,
                      const float* __restrict__ y,
                      float* __restrict__ out) {
  __shared__ __align__(16) char smem[kWgLds];

  const int lane = threadIdx.x;          // 0..31
  const int w    = threadIdx.y;          // 0..7 (one wave per row)
  const int bp   = blockIdx.x;           // 0..95  (b*P + p)
  const int n0   = blockIdx.y * (kLanes * 4) + lane * 4;   // this lane's 4 n-values

  const size_t slice = (size_t)bp * (size_t)(kC * kN);
  const float* xb = x + slice + n0;      // + c*kN per step
  const float* yb = y + slice + n0;

  char* mybuf = smem + w * kWaveLds + lane * 16;

  auto issue = [&](int step, int slot) {
    const size_t coff = (size_t)(w + kCW * step) * (size_t)kN;
    const float* gx = xb + coff;
    const float* gy = yb + coff;
    const uint32_t lx = (uint32_t)(size_t)(mybuf + slot * kSlotB);
    const uint32_t ly = lx + (uint32_t)kTileB;
    // async DMA: 32 lanes x 16B = one contiguous 512B tile per instruction
    asm volatile("global_load_async_to_lds_b128 %0, %1, off"
                 :: "v"(lx), "v"(gx) : "memory");
    asm volatile("global_load_async_to_lds_b128 %0, %1, off"
                 :: "v"(ly), "v"(gy) : "memory");
  };

  float l1x = 0.f, l1y = 0.f, l1z = 0.f, l1w = 0.f;
  float l2x = 0.f, l2y = 0.f, l2z = 0.f, l2w = 0.f;

  auto consume = [&](int slot) {
    const float4 xv = *(const float4*)(mybuf + slot * kSlotB);
    const float4 yv = *(const float4*)(mybuf + slot * kSlotB + kTileB);
    float d;
    d = xv.x - yv.x; l1x += fabsf(d); l2x = fmaf(d, d, l2x);
    d = xv.y - yv.y; l1y += fabsf(d); l2y = fmaf(d, d, l2y);
    d = xv.z - yv.z; l1z += fabsf(d); l2z = fmaf(d, d, l2z);
    d = xv.w - yv.w; l1w += fabsf(d); l2w = fmaf(d, d, l2w);
  };

  // Prologue: fill the FIFO (2 async ops per step => ASYNCcnt = 2*kDepth = 16)
#pragma unroll
  for (int s = 0; s < kDepth; ++s) issue(s, s);

  // Steady state. Completions are in-order: ASYNCcnt <= 2*(kDepth-1) means the
  // oldest (x,y) pair has landed in LDS.
#pragma unroll 8
  for (int t = 0; t < kSteps - kDepth; ++t) {
    const int slot = t & (kDepth - 1);
    asm volatile("s_wait_asynccnt 0xe" ::: "memory");
    consume(slot);
    // our ds reads of this slot must retire before the DMA overwrites it
    asm volatile("s_wait_dscnt 0x0" ::: "memory");
    issue(t + kDepth, slot);
  }

  // Epilogue: drain with a decreasing wait ladder (2*(D-1-i) remaining ops).
#define DRAIN(imm, i)                                             \
  asm volatile("s_wait_asynccnt " imm ::: "memory");              \
  consume((kSteps - kDepth + (i)) & (kDepth - 1))
  DRAIN("0xe", 0);
  DRAIN("0xc", 1);
  DRAIN("0xa", 2);
  DRAIN("0x8", 3);
  DRAIN("0x6", 4);
  DRAIN("0x4", 5);
  DRAIN("0x2", 6);
  DRAIN("0x0", 7);
#undef DRAIN

  __syncthreads();   // all waves done streaming; LDS can be reused for the reduce

  // Cross-wave reduce over the 8 c-phases, through (reused) LDS.
  float4* sL1 = (float4*)smem;                                   // [8][32]
  float4* sL2 = (float4*)(smem + kCW * kLanes * sizeof(float4)); // [8][32]
  sL1[w * kLanes + lane] = make_float4(l1x, l1y, l1z, l1w);
  sL2[w * kLanes + lane] = make_float4(l2x, l2y, l2z, l2w);
  __syncthreads();

  if (w == 0) {
#pragma unroll
    for (int r = 1; r < kCW; ++r) {
      const float4 a = sL1[r * kLanes + lane];
      const float4 s = sL2[r * kLanes + lane];
      l1x += a.x; l1y += a.y; l1z += a.z; l1w += a.w;
      l2x += s.x; l2y += s.y; l2z += s.z; l2w += s.w;
    }
    const int b = bp / kP;
    const int p = bp % kP;
    const float l1a[4] = {l1x, l1y, l1z, l1w};
    const float l2a[4] = {l2x, l2y, l2z, l2w};
#pragma unroll
    for (int j = 0; j < 4; ++j) {
      const int n = n0 + j;
      const size_t idx = (size_t)b * kOutPerB + (size_t)(p * kN + n) * 3;
      out[idx + 0] = l1a[j];
      out[idx + 1] = l2a[j];
      out[idx + 2] = l1a[j];
    }
  }
}

extern "C" void kernel_launch(void* const* d_in, const int* in_sizes, int n_in,
                              void* d_out, int out_size, void* d_ws, size_t ws_size,
                              hipStream_t stream) {
  (void)in_sizes; (void)n_in; (void)out_size; (void)d_ws; (void)ws_size;
  const float* x = (const float*)d_in[0];
  const float* y = (const float*)d_in[1];
  float* out = (float*)d_out;

  dim3 grid(kBP, kN / (kLanes * 4));   // (96, 8) = 768 workgroups
  dim3 block(kLanes, kCW);             // 256 threads = 8 wave32
  hipLaunchKernelGGL(horiz_cmp_kernel, grid, block, 0, stream, x, y, out);
}